// Decoder_49469433315870
// MI455X (gfx1250) — compile-verified
//
#include <hip/hip_runtime.h>
#include <hip/hip_bf16.h>
#include <math.h>

typedef float v2f __attribute__((ext_vector_type(2)));
typedef float v8f __attribute__((ext_vector_type(8)));

#define B_ 2048
#define S_ 200
#define V_ 50257
#define E_ 60
#define H_ 30
#define EMB_ 20

__device__ __forceinline__ float sigmoidf_(float x) { return 1.0f / (1.0f + __expf(-x)); }

// ---------------------------------------------------------------------------
// Kernel A: per-batch context, embedding gather, LSTM cell, ws_app, p_gen
// ---------------------------------------------------------------------------
__global__ __launch_bounds__(128) void kA(
    const float* __restrict__ attn, const float* __restrict__ enc_out,
    const float* __restrict__ h0, const float* __restrict__ c0,
    const int* __restrict__ dec_input, const float* __restrict__ embedding,
    const float* __restrict__ w_ih, const float* __restrict__ w_hh,
    const float* __restrict__ b_ih, const float* __restrict__ b_hh,
    const float* __restrict__ attn_ws_w, const float* __restrict__ attn_ws_b,
    const float* __restrict__ wh, const float* __restrict__ wsv, const float* __restrict__ wx,
    float* __restrict__ h_out, float* __restrict__ c_out,
    float* __restrict__ h_pad, float* __restrict__ wsapp_pad, float* __restrict__ pgen)
{
    int b = blockIdx.x;
    int tid = threadIdx.x;
    __shared__ float x[80];      // [context(60) | embed(20)]
    __shared__ float hprev[30];
    __shared__ float gates[120];
    __shared__ float hnew[32];
    __shared__ float red[128];

    if (tid < 60) {
        float acc = 0.0f;
        const float* eo = enc_out + (size_t)b * S_ * E_ + tid;
        const float* at = attn + (size_t)b * S_;
        for (int s = 0; s < S_; ++s) acc += at[s] * eo[(size_t)s * E_];
        x[tid] = acc;
    } else if (tid < 80) {
        int tok = dec_input[b];
        x[tid] = embedding[(size_t)tok * EMB_ + (tid - 60)];
    } else if (tid < 110) {
        hprev[tid - 80] = h0[b * H_ + (tid - 80)];
    }
    __syncthreads();

    if (tid < 120) {
        float g = b_ih[tid] + b_hh[tid];
        const float* wi = w_ih + tid * 80;
        #pragma unroll 8
        for (int k = 0; k < 80; ++k) g += wi[k] * x[k];
        const float* whh = w_hh + tid * H_;
        #pragma unroll 6
        for (int k = 0; k < H_; ++k) g += whh[k] * hprev[k];
        gates[tid] = g;
    }
    __syncthreads();

    if (tid < H_) {
        float ig = gates[tid], fg = gates[H_ + tid], gg = gates[2 * H_ + tid], og = gates[3 * H_ + tid];
        float c = sigmoidf_(fg) * c0[b * H_ + tid] + sigmoidf_(ig) * tanhf(gg);
        float h = sigmoidf_(og) * tanhf(c);
        c_out[b * H_ + tid] = c;
        h_out[b * H_ + tid] = h;
        hnew[tid] = h;
    } else if (tid < 32) {
        hnew[tid] = 0.0f;
    }
    __syncthreads();

    if (tid < 32) {
        h_pad[b * 32 + tid] = hnew[tid];
        float a = 0.0f;
        if (tid < H_) {
            a = attn_ws_b[tid];
            #pragma unroll 6
            for (int k = 0; k < H_; ++k) a += attn_ws_w[tid * H_ + k] * hnew[k];
        }
        wsapp_pad[b * 32 + tid] = a;
    }

    float p = 0.0f;
    if (tid < 60) p += x[tid] * wh[tid];
    if (tid >= 60 && tid < 80) p += x[tid] * wx[tid - 60];
    if (tid < H_) p += hnew[tid] * wsv[tid];
    red[tid] = p;
    __syncthreads();
    for (int off = 64; off >= 1; off >>= 1) {
        if (tid < off) red[tid] += red[tid + off];
        __syncthreads();
    }
    if (tid == 0) pgen[b] = sigmoidf_(red[0]);
}

// ---------------------------------------------------------------------------
// Kernel B: attention energy via V_WMMA_F32_16X16X4_F32 + fused tanh/dot epilogue
// rows r = b*S+s (contiguous, stride-60 in enc_out). 4 waves/block, 16 rows/wave.
// ---------------------------------------------------------------------------
__global__ __launch_bounds__(128) void kB(
    const float* __restrict__ enc_out, const float* __restrict__ coverage,
    const float* __restrict__ attn_wh_w, const float* __restrict__ attn_wh_b,
    const float* __restrict__ attn_wc, const float* __restrict__ attn_v,
    const float* __restrict__ wsapp_pad, float* __restrict__ energy)
{
    __shared__ float Wt[32 * 60];   // Wt[n*60+k] = attn_wh_w[n][k], rows 30,31 zero
    __shared__ float whb[32], wcv[32], avv[32];
    __shared__ float Ast[4 * 960];  // per-wave 16 rows x 60

    int tid = threadIdx.x;
    int wave = tid >> 5, lane = tid & 31;

    for (int i = tid; i < 32 * 60; i += 128) {
        int n = i / 60, k = i % 60;
        Wt[i] = (n < H_) ? attn_wh_w[n * E_ + k] : 0.0f;
    }
    if (tid < 32) {
        whb[tid] = (tid < H_) ? attn_wh_b[tid] : 0.0f;
        wcv[tid] = (tid < H_) ? attn_wc[tid] : 0.0f;
        avv[tid] = (tid < H_) ? attn_v[tid] : 0.0f;
    }
    int r0 = blockIdx.x * 64 + wave * 16;
    float* Al = &Ast[wave * 960];
    const float* src = enc_out + (size_t)r0 * E_;
    for (int i = lane; i < 960; i += 32) Al[i] = src[i];
    __syncthreads();

    int m = lane & 15;
    int k0 = (lane >> 4) * 2;
    int ncol0 = m, ncol1 = 16 + m;

    v8f acc0 = {}, acc1 = {};
    #pragma unroll
    for (int kk = 0; kk < 15; ++kk) {
        int k = kk * 4 + k0;
        v2f a;  a.x  = Al[m * 60 + k];      a.y  = Al[m * 60 + k + 1];
        v2f b0; b0.x = Wt[ncol0 * 60 + k];  b0.y = Wt[ncol0 * 60 + k + 1];
        v2f b1; b1.x = Wt[ncol1 * 60 + k];  b1.y = Wt[ncol1 * 60 + k + 1];
        acc0 = __builtin_amdgcn_wmma_f32_16x16x4_f32(false, a, false, b0, (short)0, acc0, false, false);
        acc1 = __builtin_amdgcn_wmma_f32_16x16x4_f32(false, a, false, b1, (short)0, acc1, false, false);
    }

    int hi = lane >> 4;
    float whb0 = whb[ncol0], whb1 = whb[ncol1];
    float wc0 = wcv[ncol0],  wc1 = wcv[ncol1];
    float av0 = avv[ncol0],  av1 = avv[ncol1];

    #pragma unroll
    for (int vv = 0; vv < 8; ++vv) {
        int r = r0 + vv + 8 * hi;
        int bb = r / S_;
        float cov = coverage[r];
        float ws0 = wsapp_pad[bb * 32 + ncol0];
        float ws1 = wsapp_pad[bb * 32 + ncol1];
        float t = tanhf(acc0[vv] + whb0 + ws0 + cov * wc0) * av0
                + tanhf(acc1[vv] + whb1 + ws1 + cov * wc1) * av1;
        // reduce over N (16 lanes per half-wave)
        for (int off = 8; off >= 1; off >>= 1) t += __shfl_xor(t, off, 32);
        if (m == 0) energy[r] = t;
    }
}

// ---------------------------------------------------------------------------
// Kernel C: softmax over S, new_coverage, per-batch coverage_loss partial
// ---------------------------------------------------------------------------
__global__ __launch_bounds__(256) void kC(
    const float* __restrict__ energy, const float* __restrict__ coverage,
    float* __restrict__ new_attn, float* __restrict__ new_cov, float* __restrict__ partial)
{
    int b = blockIdx.x, tid = threadIdx.x;
    __shared__ float red[256];
    float e = (tid < S_) ? energy[b * S_ + tid] : -3.4e38f;
    red[tid] = e; __syncthreads();
    for (int off = 128; off >= 1; off >>= 1) {
        if (tid < off) red[tid] = fmaxf(red[tid], red[tid + off]);
        __syncthreads();
    }
    float mx = red[0]; __syncthreads();
    float ex = (tid < S_) ? __expf(e - mx) : 0.0f;
    red[tid] = ex; __syncthreads();
    for (int off = 128; off >= 1; off >>= 1) {
        if (tid < off) red[tid] += red[tid + off];
        __syncthreads();
    }
    float sum = red[0]; __syncthreads();
    float pl = 0.0f;
    if (tid < S_) {
        float na = ex / sum;
        float cv = coverage[b * S_ + tid];
        new_attn[b * S_ + tid] = na;
        new_cov[b * S_ + tid] = cv + na;
        pl = fminf(na, cv);
    }
    red[tid] = pl; __syncthreads();
    for (int off = 128; off >= 1; off >>= 1) {
        if (tid < off) red[tid] += red[tid + off];
        __syncthreads();
    }
    if (tid == 0) partial[b] = red[0];
}

__global__ __launch_bounds__(256) void kC2(const float* __restrict__ partial, float* __restrict__ loss)
{
    int tid = threadIdx.x;
    __shared__ float red[256];
    float s = 0.0f;
    for (int i = tid; i < B_; i += 256) s += partial[i];
    red[tid] = s; __syncthreads();
    for (int off = 128; off >= 1; off >>= 1) {
        if (tid < off) red[tid] += red[tid + off];
        __syncthreads();
    }
    if (tid == 0) loss[0] = red[0];
}

// ---------------------------------------------------------------------------
// Kernel D: logits = h @ v_w^T + v_b via V_WMMA_F32_16X16X4_F32
// grid = (vocab chunks of 256, batch tiles of 16); 4 waves, each 16x64.
// ---------------------------------------------------------------------------
__global__ __launch_bounds__(128) void kD(
    const float* __restrict__ h_pad, const float* __restrict__ v_w,
    const float* __restrict__ v_b, float* __restrict__ out)
{
    __shared__ float Bl[256 * 34];   // padded stride 34: bank-conflict-free b64 reads
    __shared__ float vb[256];

    int tid = threadIdx.x;
    int wave = tid >> 5, lane = tid & 31;
    int b0 = blockIdx.y * 16;
    int v0 = blockIdx.x * 256;

    for (int i = tid; i < 256 * 32; i += 128) {
        int vr = i >> 5, k = i & 31;
        int v = v0 + vr;
        Bl[vr * 34 + k] = (k < H_ && v < V_) ? v_w[(size_t)v * H_ + k] : 0.0f;
    }
    for (int i = tid; i < 256; i += 128) vb[i] = (v0 + i < V_) ? v_b[v0 + i] : 0.0f;
    __syncthreads();

    int m = lane & 15, k0 = (lane >> 4) * 2, hi = lane >> 4;

    v2f A[8];
    const float* hp = h_pad + (size_t)(b0 + m) * 32;
    #pragma unroll
    for (int kk = 0; kk < 8; ++kk) { A[kk].x = hp[kk * 4 + k0]; A[kk].y = hp[kk * 4 + k0 + 1]; }

    #pragma unroll
    for (int t = 0; t < 4; ++t) {
        int nloc = wave * 64 + t * 16 + m;
        v8f acc = {};
        #pragma unroll
        for (int kk = 0; kk < 8; ++kk) {
            v2f bf; bf.x = Bl[nloc * 34 + kk * 4 + k0]; bf.y = Bl[nloc * 34 + kk * 4 + k0 + 1];
            acc = __builtin_amdgcn_wmma_f32_16x16x4_f32(false, A[kk], false, bf, (short)0, acc, false, false);
        }
        int col = v0 + nloc;
        if (col < V_) {
            float bias = vb[nloc];
            #pragma unroll
            for (int vv = 0; vv < 8; ++vv) {
                int row = b0 + vv + 8 * hi;
                out[(size_t)row * V_ + col] = acc[vv] + bias;
            }
        }
    }
}

// ---------------------------------------------------------------------------
// Kernel E: per-row online max/sum-of-exp over V
// ---------------------------------------------------------------------------
__global__ __launch_bounds__(256) void kE(
    const float* __restrict__ logits, float* __restrict__ rowmax, float* __restrict__ rowsum)
{
    int b = blockIdx.x, tid = threadIdx.x;
    const float* row = logits + (size_t)b * V_;
    float m = -3.4e38f, s = 0.0f;
    for (int j = tid; j < V_; j += 256) {
        float x = row[j];
        if (x <= m) s += __expf(x - m);
        else { s = s * __expf(m - x) + 1.0f; m = x; }
    }
    __shared__ float sm[256], ss[256];
    sm[tid] = m; ss[tid] = s; __syncthreads();
    for (int off = 128; off >= 1; off >>= 1) {
        if (tid < off) {
            float m2 = sm[tid + off], s2 = ss[tid + off];
            float nm = fmaxf(sm[tid], m2);
            ss[tid] = ss[tid] * __expf(sm[tid] - nm) + s2 * __expf(m2 - nm);
            sm[tid] = nm;
        }
        __syncthreads();
    }
    if (tid == 0) { rowmax[b] = sm[0]; rowsum[b] = ss[0]; }
}

// ---------------------------------------------------------------------------
// Kernel G1: precompute scattered values (reads raw logits BEFORE normalize)
// ---------------------------------------------------------------------------
__global__ __launch_bounds__(256) void kG1(
    const int* __restrict__ enc_inputs, const float* __restrict__ logits,
    const float* __restrict__ rowmax, const float* __restrict__ rowsum,
    const float* __restrict__ pgen, const float* __restrict__ new_attn,
    float* __restrict__ scval)
{
    int i = blockIdx.x * 256 + threadIdx.x;
    if (i >= B_ * S_) return;
    int b = i / S_;
    int col = enc_inputs[i];
    float l = logits[(size_t)b * V_ + col];
    float pv = __expf(l - rowmax[b]) / rowsum[b];
    float pg = pgen[b];
    scval[i] = pg * pv + (1.0f - pg) * new_attn[i];
}

// ---------------------------------------------------------------------------
// Kernel F: in-place normalize + p_gen blend of the generation distribution
// ---------------------------------------------------------------------------
__global__ __launch_bounds__(256) void kF(
    float* __restrict__ out, const float* __restrict__ rowmax,
    const float* __restrict__ rowsum, const float* __restrict__ pgen)
{
    int b = blockIdx.y;
    int j = blockIdx.x * 256 + threadIdx.x;
    if (j >= V_) return;
    size_t idx = (size_t)b * V_ + j;
    out[idx] = pgen[b] * __expf(out[idx] - rowmax[b]) / rowsum[b];
}

// ---------------------------------------------------------------------------
// Kernel H: pointer scatter (last-wins store matches .at[].set duplicate UB)
// ---------------------------------------------------------------------------
__global__ __launch_bounds__(256) void kH(
    const int* __restrict__ enc_inputs, const float* __restrict__ scval, float* __restrict__ out)
{
    int i = blockIdx.x * 256 + threadIdx.x;
    if (i >= B_ * S_) return;
    int b = i / S_;
    int col = enc_inputs[i];
    out[(size_t)b * V_ + col] = scval[i];
}

// ---------------------------------------------------------------------------
extern "C" void kernel_launch(void* const* d_in, const int* in_sizes, int n_in,
                              void* d_out, int out_size, void* d_ws, size_t ws_size,
                              hipStream_t stream)
{
    (void)in_sizes; (void)n_in; (void)out_size; (void)ws_size;

    const float* coverage   = (const float*)d_in[0];
    const float* enc_out    = (const float*)d_in[1];
    const float* h0         = (const float*)d_in[2];
    const float* c0         = (const float*)d_in[3];
    const float* attn       = (const float*)d_in[4];
    const int*   dec_input  = (const int*)d_in[5];
    const int*   enc_inputs = (const int*)d_in[6];
    const float* embedding  = (const float*)d_in[7];
    const float* w_ih       = (const float*)d_in[8];
    const float* w_hh       = (const float*)d_in[9];
    const float* b_ih       = (const float*)d_in[10];
    const float* b_hh       = (const float*)d_in[11];
    const float* attn_wh_w  = (const float*)d_in[12];
    const float* attn_wh_b  = (const float*)d_in[13];
    const float* attn_ws_w  = (const float*)d_in[14];
    const float* attn_ws_b  = (const float*)d_in[15];
    const float* attn_wc    = (const float*)d_in[16];
    const float* attn_v     = (const float*)d_in[17];
    const float* wh         = (const float*)d_in[18];
    const float* wsv        = (const float*)d_in[19];
    const float* wx         = (const float*)d_in[20];
    const float* v_w        = (const float*)d_in[21];
    const float* v_b        = (const float*)d_in[22];

    float* out = (float*)d_out;
    // output layout: [output BxV | new_coverage BxS | h BxH | c BxH | new_attn BxS | loss 1]
    const size_t OUT_OFF  = 0;
    const size_t NC_OFF   = (size_t)B_ * V_;
    const size_t H_OFF    = NC_OFF + (size_t)B_ * S_;
    const size_t C_OFF    = H_OFF + (size_t)B_ * H_;
    const size_t NA_OFF   = C_OFF + (size_t)B_ * H_;
    const size_t LOSS_OFF = NA_OFF + (size_t)B_ * S_;

    // workspace layout (floats)
    float* w = (float*)d_ws;
    float* ws_hpad    = w;                          // B*32
    float* ws_wsapp   = ws_hpad  + (size_t)B_ * 32; // B*32
    float* ws_pgen    = ws_wsapp + (size_t)B_ * 32; // B
    float* ws_energy  = ws_pgen  + B_;              // B*S
    float* ws_rowmax  = ws_energy + (size_t)B_ * S_;// B
    float* ws_rowsum  = ws_rowmax + B_;             // B
    float* ws_partial = ws_rowsum + B_;             // B
    float* ws_scval   = ws_partial + B_;            // B*S

    // A: context + LSTM + ws_app + p_gen
    kA<<<B_, 128, 0, stream>>>(attn, enc_out, h0, c0, dec_input, embedding,
                               w_ih, w_hh, b_ih, b_hh, attn_ws_w, attn_ws_b,
                               wh, wsv, wx,
                               out + H_OFF, out + C_OFF, ws_hpad, ws_wsapp, ws_pgen);

    // B: attention energy (WMMA f32 16x16x4)
    kB<<<(B_ * S_) / 64, 128, 0, stream>>>(enc_out, coverage, attn_wh_w, attn_wh_b,
                                           attn_wc, attn_v, ws_wsapp, ws_energy);

    // C: softmax over S, new_coverage, loss partials; C2 reduce
    kC<<<B_, 256, 0, stream>>>(ws_energy, coverage, out + NA_OFF, out + NC_OFF, ws_partial);
    kC2<<<1, 256, 0, stream>>>(ws_partial, out + LOSS_OFF);

    // D: vocab logits (WMMA f32 16x16x4) into out[0 .. B*V)
    dim3 gD((V_ + 255) / 256, B_ / 16);
    kD<<<gD, 128, 0, stream>>>(ws_hpad, v_w, v_b, out + OUT_OFF);

    // E: per-row softmax stats over V
    kE<<<B_, 256, 0, stream>>>(out + OUT_OFF, ws_rowmax, ws_rowsum);

    // G1: precompute scattered blend values from raw logits
    kG1<<<(B_ * S_ + 255) / 256, 256, 0, stream>>>(enc_inputs, out + OUT_OFF, ws_rowmax,
                                                   ws_rowsum, ws_pgen, out + NA_OFF, ws_scval);

    // F: in-place normalize + p_gen scaling
    dim3 gF((V_ + 255) / 256, B_);
    kF<<<gF, 256, 0, stream>>>(out + OUT_OFF, ws_rowmax, ws_rowsum, ws_pgen);

    // H: pointer scatter
    kH<<<(B_ * S_ + 255) / 256, 256, 0, stream>>>(enc_inputs, ws_scval, out + OUT_OFF);
}